// MOFTCrossAttnProcessor_74217034875000
// MI455X (gfx1250) — compile-verified
//
#include <hip/hip_runtime.h>

typedef __attribute__((ext_vector_type(2))) float v2f;
typedef __attribute__((ext_vector_type(8))) float v8f;

__device__ inline v8f wmma_f32(v2f a, v2f b, v8f c) {
  // D = A(16x4) * B(4x16) + C, all f32, full precision
  return __builtin_amdgcn_wmma_f32_16x16x4_f32(false, a, false, b, (short)0, c,
                                               false, false);
}

// ---------------------------------------------------------------------------
// Cayley: X = (I+S)^-1 (I-S), S = W - W^T, per-matrix Gauss-Jordan in LDS.
// One workgroup per matrix, n <= 80.
// ---------------------------------------------------------------------------
__global__ __launch_bounds__(256) void cayley_kernel(const float* __restrict__ W,
                                                     float* __restrict__ X, int n) {
  __shared__ float sM[6400];
  __shared__ float sX[6400];
  __shared__ float fac[80];
  const int nn = n * n;
  const float* Wm = W + (size_t)blockIdx.x * nn;
  float* Xo = X + (size_t)blockIdx.x * nn;
  const int tid = threadIdx.x;
  for (int idx = tid; idx < nn; idx += 256) {
    int i = idx / n, j = idx - i * n;
    float s = Wm[i * n + j] - Wm[j * n + i];
    float d = (i == j) ? 1.0f : 0.0f;
    sM[idx] = d + s;
    sX[idx] = d - s;
  }
  __syncthreads();
  for (int p = 0; p < n; ++p) {
    float inv = 1.0f / sM[p * n + p];
    __syncthreads();
    for (int j = tid; j < n; j += 256) {
      sM[p * n + j] *= inv;
      sX[p * n + j] *= inv;
    }
    __syncthreads();
    for (int r = tid; r < n; r += 256) fac[r] = (r == p) ? 0.0f : sM[r * n + p];
    __syncthreads();
    for (int idx = tid; idx < nn; idx += 256) {
      int r = idx / n;
      if (r == p) continue;
      int j = idx - r * n;
      float f = fac[r];
      sM[idx] -= f * sM[p * n + j];
      sX[idx] -= f * sX[p * n + j];
    }
    __syncthreads();
  }
  for (int idx = tid; idx < nn; idx += 256) Xo[idx] = sX[idx];
}

// ---------------------------------------------------------------------------
// Combined monarch matrix: Mo[k*b+b0, j*b+c] = A[k,b0,c] * B[c,k,j]
// (monarch(x) == x @ Mo).  Pure elementwise product.
// ---------------------------------------------------------------------------
__global__ void build_monarch(const float* __restrict__ A, const float* __restrict__ Bm,
                              float* __restrict__ Mo, int nb, int b) {
  int n = nb * b;
  int idx = blockIdx.x * blockDim.x + threadIdx.x;
  if (idx >= n * n) return;
  int row = idx / n, col = idx - row * n;
  int k = row / b, b0 = row - k * b;
  int j = col / b, c = col - j * b;
  Mo[idx] = A[((size_t)k * b + b0) * b + c] * Bm[((size_t)c * nb + k) * nb + j];
}

// ---------------------------------------------------------------------------
// f32 WMMA GEMM (NT):  C[M,N] = A[M,K] @ B[N,K]^T
// Optional: rowScale[m] multiplies row m of C; colBias[n] added to column n.
// Block tile 128x64x16, 8 waves as 4(m) x 2(n), each wave 32x32 (2x2 v8f accs)
// -> 1.0 LDS fragment loads per wmma.  Ping-pong LDS double buffering with a
// single barrier per K-tile; next tile's global loads overlap the WMMA burst.
// ---------------------------------------------------------------------------
#define GS 17  // LDS row stride (17 coprime with 64 banks)
__global__ __launch_bounds__(256) void gemm_nt(const float* __restrict__ A,
                                               const float* __restrict__ B,
                                               float* __restrict__ C, int M, int N,
                                               int K, const float* __restrict__ rowScale,
                                               const float* __restrict__ colBias) {
  __shared__ float sA[2][128 * GS];
  __shared__ float sB[2][64 * GS];
  const int n0 = blockIdx.x * 64;
  const int m0 = blockIdx.y * 128;
  const int tid = threadIdx.x;
  const int lane = tid & 31;
  const int wave = tid >> 5;
  const int moff = (wave & 3) * 32;  // 0,32,64,96
  const int noff = (wave >> 2) * 32; // 0,32
  const int lr = lane & 15;
  const int kb = (lane >> 4) << 1;
  v8f c00 = {}, c01 = {}, c10 = {}, c11 = {};
  // staging thread mapping
  const int arow = tid >> 1;       // 0..127
  const int acol = (tid & 1) << 3; // 0 or 8
  const int brow = tid >> 2;       // 0..63
  const int bcol = (tid & 3) << 2; // 0,4,8,12
  float ra[8], rb[4];

  auto gload = [&](int k0) {
    int gm = m0 + arow;
    const float* ap = A + (size_t)gm * K + k0 + acol;
#pragma unroll
    for (int j = 0; j < 8; ++j) ra[j] = (gm < M) ? ap[j] : 0.f;
    int gn = n0 + brow;
    const float* bp = B + (size_t)gn * K + k0 + bcol;
#pragma unroll
    for (int j = 0; j < 4; ++j) rb[j] = (gn < N) ? bp[j] : 0.f;
  };

  gload(0);
  int p = 0;
  for (int k0 = 0; k0 < K; k0 += 16) {
#pragma unroll
    for (int j = 0; j < 8; ++j) sA[p][arow * GS + acol + j] = ra[j];
#pragma unroll
    for (int j = 0; j < 4; ++j) sB[p][brow * GS + bcol + j] = rb[j];
    if (k0 + 16 < K) gload(k0 + 16);
    __syncthreads();
    const float* pa = &sA[p][0];
    const float* pb = &sB[p][0];
#pragma unroll
    for (int kk = 0; kk < 16; kk += 4) {
      v2f a0, a1, b0, b1;
      a0.x = pa[(moff + lr) * GS + kk + kb];
      a0.y = pa[(moff + lr) * GS + kk + kb + 1];
      a1.x = pa[(moff + 16 + lr) * GS + kk + kb];
      a1.y = pa[(moff + 16 + lr) * GS + kk + kb + 1];
      b0.x = pb[(noff + lr) * GS + kk + kb];
      b0.y = pb[(noff + lr) * GS + kk + kb + 1];
      b1.x = pb[(noff + 16 + lr) * GS + kk + kb];
      b1.y = pb[(noff + 16 + lr) * GS + kk + kb + 1];
      c00 = wmma_f32(a0, b0, c00);
      c01 = wmma_f32(a0, b1, c01);
      c10 = wmma_f32(a1, b0, c10);
      c11 = wmma_f32(a1, b1, c11);
    }
    p ^= 1;
  }
  const int rhalf = (lane >> 4) << 3;
  const int gn0 = n0 + noff + lr;
#pragma unroll
  for (int mi = 0; mi < 2; ++mi) {
    v8f* cm0 = mi ? &c10 : &c00;
    v8f* cm1 = mi ? &c11 : &c01;
#pragma unroll
    for (int i = 0; i < 8; ++i) {
      int gm = m0 + moff + mi * 16 + i + rhalf;
      if (gm >= M) continue;
      float rs = rowScale ? rowScale[gm] : 1.0f;
      float v0 = (*cm0)[i] * rs;
      float v1 = (*cm1)[i] * rs;
      if (colBias) { v0 += colBias[gn0]; v1 += colBias[gn0 + 16]; }
      C[(size_t)gm * N + gn0] = v0;
      C[(size_t)gm * N + gn0 + 16] = v1;
    }
  }
}

// ---------------------------------------------------------------------------
// Attention: per (b,h) full K/V (T=77, dh=64) in LDS; each block = 4 waves,
// 64 query rows; WMMA scores -> exact masked softmax -> WMMA P@V.
// Q is read from d_out (written by the Q projection GEMM).
// ---------------------------------------------------------------------------
__global__ __launch_bounds__(128) void attn_kernel(const float* __restrict__ Q,
                                                   const float* __restrict__ Kb,
                                                   const float* __restrict__ Vb,
                                                   float* __restrict__ O) {
  __shared__ float sK[80 * 65];
  __shared__ float sV[80 * 65];
  __shared__ float sP[4][16 * 81];
  const int bh = blockIdx.y;
  const int b = bh / 20, h = bh - b * 20;
  const int q0 = blockIdx.x * 64;
  const int tid = threadIdx.x;
  const int lane = tid & 31, wave = tid >> 5;
  for (int idx = tid; idx < 80 * 64; idx += 128) {
    int t = idx >> 6, d = idx & 63;
    float kv = 0.f, vv = 0.f;
    if (t < 77) {
      size_t r = (size_t)(b * 77 + t) * 1280 + h * 64 + d;
      kv = Kb[r];
      vv = Vb[r];
    }
    sK[t * 65 + d] = kv;
    sV[t * 65 + d] = vv;
  }
  __syncthreads();
  const int lr = lane & 15;
  const int kb = (lane >> 4) << 1;
  const int qrow = b * 4096 + q0 + wave * 16;
  v8f acc[5] = {};
  const float* qp = Q + (size_t)(qrow + lr) * 1280 + h * 64;
  for (int kk = 0; kk < 64; kk += 4) {
    v2f a;
    a.x = qp[kk + kb];
    a.y = qp[kk + kb + 1];
#pragma unroll
    for (int t = 0; t < 5; ++t) {
      v2f bf;
      bf.x = sK[(t * 16 + lr) * 65 + kk + kb];
      bf.y = sK[(t * 16 + lr) * 65 + kk + kb + 1];
      acc[t] = wmma_f32(a, bf, acc[t]);
    }
  }
  const float scl = 0.125f;  // 64^-0.5
  const int rhalf = (lane >> 4) << 3;
#pragma unroll
  for (int i = 0; i < 8; ++i) {
    float e[5];
    float m = -3.4e38f;
#pragma unroll
    for (int t = 0; t < 5; ++t) {
      int col = t * 16 + lr;
      float v = acc[t][i] * scl;
      if (col >= 77) v = -3.4e38f;
      e[t] = v;
      m = fmaxf(m, v);
    }
    for (int off = 1; off < 16; off <<= 1) m = fmaxf(m, __shfl_xor(m, off, 32));
    float s = 0.f;
#pragma unroll
    for (int t = 0; t < 5; ++t) {
      int col = t * 16 + lr;
      float pv = (col < 77) ? __expf(e[t] - m) : 0.f;
      e[t] = pv;
      s += pv;
    }
    for (int off = 1; off < 16; off <<= 1) s += __shfl_xor(s, off, 32);
    float rinv = 1.0f / s;
    int prow = i + rhalf;
#pragma unroll
    for (int t = 0; t < 5; ++t) sP[wave][prow * 81 + t * 16 + lr] = e[t] * rinv;
  }
  __syncthreads();
  v8f o[4] = {};
  for (int kk = 0; kk < 80; kk += 4) {
    v2f a;
    a.x = sP[wave][lr * 81 + kk + kb];
    a.y = sP[wave][lr * 81 + kk + kb + 1];
#pragma unroll
    for (int t = 0; t < 4; ++t) {
      v2f bf;
      bf.x = sV[(kk + kb) * 65 + t * 16 + lr];
      bf.y = sV[(kk + kb + 1) * 65 + t * 16 + lr];
      o[t] = wmma_f32(a, bf, o[t]);
    }
  }
#pragma unroll
  for (int i = 0; i < 8; ++i) {
    int gr = qrow + i + rhalf;
    float* op = O + (size_t)gr * 1280 + h * 64;
#pragma unroll
    for (int t = 0; t < 4; ++t) op[t * 16 + lr] = o[t][i];
  }
}

// ---------------------------------------------------------------------------
extern "C" void kernel_launch(void* const* d_in, const int* in_sizes, int n_in,
                              void* d_out, int out_size, void* d_ws, size_t ws_size,
                              hipStream_t stream) {
  (void)in_sizes; (void)n_in; (void)out_size; (void)ws_size;
  const float* hidden = (const float*)d_in[0];
  const float* enc = (const float*)d_in[1];
  const float* qA = (const float*)d_in[2];
  const float* qB = (const float*)d_in[3];
  const float* kA = (const float*)d_in[4];
  const float* kB = (const float*)d_in[5];
  const float* vA = (const float*)d_in[6];
  const float* vB = (const float*)d_in[7];
  const float* oA = (const float*)d_in[8];
  const float* oB = (const float*)d_in[9];
  const float* Wq = (const float*)d_in[10];
  const float* Wk = (const float*)d_in[11];
  const float* Wv = (const float*)d_in[12];
  const float* Wo = (const float*)d_in[13];
  const float* bo = (const float*)d_in[14];
  const float* qs = (const float*)d_in[15];
  const float* ks = (const float*)d_in[16];
  const float* vs = (const float*)d_in[17];
  const float* os = (const float*)d_in[18];
  float* out = (float*)d_out;
  float* ws = (float*)d_ws;

  const int B = 8, S = 4096, D = 1280, E = 768, T = 77, Hh = 20;
  size_t off = 0;
  auto take = [&](size_t n) { float* p = ws + off; off += n; return p; };
  float* Aq = take((size_t)16 * 80 * 80);
  float* Ak = take((size_t)16 * 48 * 48);
  float* Av = take((size_t)16 * 48 * 48);
  float* Ao = take((size_t)16 * 80 * 80);
  float* Bq = take((size_t)80 * 16 * 16);
  float* Bk = take((size_t)48 * 16 * 16);
  float* Bv = take((size_t)48 * 16 * 16);
  float* Bo = take((size_t)80 * 16 * 16);
  float* Mq = take((size_t)D * D);
  float* Mk = take((size_t)E * E);
  float* Mv = take((size_t)E * E);
  float* Mo = take((size_t)D * D);
  float* WTq = take((size_t)D * D);
  float* WTk = take((size_t)D * E);
  float* WTv = take((size_t)D * E);
  float* WTo = take((size_t)D * D);
  float* Kb = take((size_t)B * T * D);
  float* Vb = take((size_t)B * T * D);
  float* At = take((size_t)B * S * D);

  // 1) Cayley orthogonalization of all monarch blocks.
  cayley_kernel<<<16, 256, 0, stream>>>(qA, Aq, 80);
  cayley_kernel<<<16, 256, 0, stream>>>(kA, Ak, 48);
  cayley_kernel<<<16, 256, 0, stream>>>(vA, Av, 48);
  cayley_kernel<<<16, 256, 0, stream>>>(oA, Ao, 80);
  cayley_kernel<<<80, 256, 0, stream>>>(qB, Bq, 16);
  cayley_kernel<<<48, 256, 0, stream>>>(kB, Bk, 16);
  cayley_kernel<<<48, 256, 0, stream>>>(vB, Bv, 16);
  cayley_kernel<<<80, 256, 0, stream>>>(oB, Bo, 16);

  // 2) Combined monarch matrices (elementwise).
  build_monarch<<<(D * D + 255) / 256, 256, 0, stream>>>(Aq, Bq, Mq, 16, 80);
  build_monarch<<<(E * E + 255) / 256, 256, 0, stream>>>(Ak, Bk, Mk, 16, 48);
  build_monarch<<<(E * E + 255) / 256, 256, 0, stream>>>(Av, Bv, Mv, 16, 48);
  build_monarch<<<(D * D + 255) / 256, 256, 0, stream>>>(Ao, Bo, Mo, 16, 80);

  // 3) Effective weights (transposed):  WT[d,i] = scale[d] * sum_m W[d,m]*M[i,m]
  gemm_nt<<<dim3(D / 64, D / 128), 256, 0, stream>>>(Wq, Mq, WTq, D, D, D, qs, nullptr);
  gemm_nt<<<dim3(E / 64, D / 128), 256, 0, stream>>>(Wk, Mk, WTk, D, E, E, ks, nullptr);
  gemm_nt<<<dim3(E / 64, D / 128), 256, 0, stream>>>(Wv, Mv, WTv, D, E, E, vs, nullptr);
  gemm_nt<<<dim3(D / 64, D / 128), 256, 0, stream>>>(Wo, Mo, WTo, D, D, D, os, nullptr);

  // 4) Projections. Q lives in d_out as scratch (fully overwritten at the end).
  gemm_nt<<<dim3(D / 64, (B * S) / 128), 256, 0, stream>>>(hidden, WTq, out, B * S, D,
                                                           D, nullptr, nullptr);
  gemm_nt<<<dim3(D / 64, (B * T + 127) / 128), 256, 0, stream>>>(enc, WTk, Kb, B * T,
                                                                 D, E, nullptr, nullptr);
  gemm_nt<<<dim3(D / 64, (B * T + 127) / 128), 256, 0, stream>>>(enc, WTv, Vb, B * T,
                                                                 D, E, nullptr, nullptr);

  // 5) Attention (exact softmax over T=77).
  attn_kernel<<<dim3(S / 64, B * Hh), 128, 0, stream>>>(out, Kb, Vb, At);

  // 6) Output projection + bias -> d_out.
  gemm_nt<<<dim3(D / 64, (B * S) / 128), 256, 0, stream>>>(At, WTo, out, B * S, D, D,
                                                           nullptr, bo);
}